// HVRTLinearFFN_75883482186212
// MI455X (gfx1250) — compile-verified
//
#include <hip/hip_runtime.h>

// Problem constants (match reference)
#define P_PART 64
#define DIN    1024
#define RANK   64
#define DOUT   1024
#define NTOK   4096
#define MAX_TILES (NTOK / 16 + P_PART)   // sum ceil(cnt_p/16) <= N/16 + P = 320

typedef __attribute__((ext_vector_type(2))) float v2f;
typedef __attribute__((ext_vector_type(4))) float v4f;
typedef __attribute__((ext_vector_type(8))) float v8f;

// ---------------- prepass: bucket tokens by pid ----------------

__global__ void k_zero(int* counts, int* cursor) {
    int i = threadIdx.x;
    if (i < P_PART) { counts[i] = 0; cursor[i] = 0; }
}

__global__ void k_count(const int* __restrict__ pids, int* __restrict__ counts) {
    int i = blockIdx.x * blockDim.x + threadIdx.x;
    if (i < NTOK) atomicAdd(&counts[pids[i]], 1);
}

// Serial over P=64: exclusive scan -> offsets, plus build compact tile worklist.
__global__ void k_scan(const int* __restrict__ counts, int* __restrict__ offsets,
                       int* __restrict__ tiles, int* __restrict__ ntiles) {
    if (threadIdx.x == 0) {
        int s = 0, t = 0;
        for (int p = 0; p < P_PART; ++p) {
            offsets[p] = s;
            int c = counts[p];
            s += c;
            for (int t0 = 0; t0 < c; t0 += 16)
                tiles[t++] = (p << 16) | t0;   // pack (partition, tile start)
        }
        offsets[P_PART] = s;
        *ntiles = t;
    }
}

__global__ void k_scatter(const int* __restrict__ pids, const int* __restrict__ offsets,
                          int* __restrict__ cursor, int* __restrict__ order) {
    int i = blockIdx.x * blockDim.x + threadIdx.x;
    if (i < NTOK) {
        int p = pids[i];
        int pos = offsets[p] + atomicAdd(&cursor[p], 1);
        order[pos] = i;
    }
}

// ---------------- grouped low-rank GEMM via FP32 WMMA ----------------
// Block = 128 threads (4 waves). One block = one 16-token tile of one partition.
// Phase 1: H[16x64] = Xtile[16x1024] @ U_p[1024x64]   (wave w -> H columns 16w..16w+15)
// Phase 2: Out[16x1024] = H @ V_p[64x1024] + bias     (wave w -> D_OUT cols 256w..256w+255)

#define XS_STRIDE 1028   // 1024 + 4 pad floats: bank = (4*row + col) % 64 -> conflict-free
#define HS_STRIDE 68     // 64 + 4 pad floats

__global__ __launch_bounds__(128) void grouped_lowrank_gemm(
    const float* __restrict__ x,
    const float* __restrict__ U,
    const float* __restrict__ V,
    const float* __restrict__ bias,
    const int*  __restrict__ offsets,
    const int*  __restrict__ order,
    const int*  __restrict__ tiles,
    const int*  __restrict__ ntiles,
    float* __restrict__ out)
{
    __shared__ float xs[16 * XS_STRIDE];   // 65792 B
    __shared__ float hs[16 * HS_STRIDE];   // 4352 B
    __shared__ int   toks[16];

    if ((int)blockIdx.x >= *ntiles) return;   // compact worklist: ~320 live blocks total
    const int ent  = tiles[blockIdx.x];
    const int p    = ent >> 16;
    const int t0   = ent & 0xffff;
    const int base = offsets[p];
    const int cnt  = offsets[p + 1] - base;

    const int tid = threadIdx.x;

    if (tid < 16) {
        int idx = t0 + tid;
        int clamped = (idx < cnt) ? idx : (cnt - 1);  // pad partial tile with a valid row
        toks[tid] = order[base + clamped];
    }
    __syncthreads();

    // ---- stage gathered X tile into LDS (128B bursts per 8-lane group) ----
    {
        int row   = tid >> 3;        // 0..15
        int lane8 = tid & 7;
        const float* xrow = x + (size_t)toks[row] * DIN;
        for (int c = lane8 * 4; c < DIN; c += 32) {
            v4f v = *(const v4f*)(xrow + c);
            *(v4f*)(&xs[row * XS_STRIDE + c]) = v;
        }
    }
    __syncthreads();

    const int wave  = tid >> 5;      // 0..3
    const int lane  = tid & 31;
    const int lhalf = lane >> 4;     // K-half selector per 32-bit WMMA operand layout
    const int l16   = lane & 15;     // M (for A) / N (for B,C,D)

    // ---- phase 1: H tile (16 tokens x 16 rank cols), K = 1024 ----
    {
        const float* Up = U + (size_t)p * DIN * RANK;
        const int ncol = wave * 16 + l16;          // rank column this lane produces
        v8f acc = {};
        for (int kb = 0; kb < DIN / 4; ++kb) {
            int k = kb * 4 + lhalf * 2;
            v2f a = *(const v2f*)(&xs[l16 * XS_STRIDE + k]);   // A: rows = tokens
            v2f b;
            b.x = Up[(size_t)k       * RANK + ncol];           // B: U rows k, k+1
            b.y = Up[(size_t)(k + 1) * RANK + ncol];
            acc = __builtin_amdgcn_wmma_f32_16x16x4_f32(
                false, a, false, b, (short)0, acc, false, false);
        }
        // C/D layout: VGPR j holds M = j + 8*lhalf, N = l16
        #pragma unroll
        for (int j = 0; j < 8; ++j)
            hs[(j + 8 * lhalf) * HS_STRIDE + ncol] = acc[j];
    }
    __syncthreads();

    // ---- phase 2: Out slice (16 tokens x 256 cols), K = 64 ----
    {
        const float* Vp   = V + (size_t)p * RANK * DOUT;
        const int   nbase = wave * 256;
        for (int nt = 0; nt < 16; ++nt) {
            const int ncol = nbase + nt * 16 + l16;
            v8f acc = {};
            for (int kb = 0; kb < RANK / 4; ++kb) {
                int k = kb * 4 + lhalf * 2;
                v2f a = *(const v2f*)(&hs[l16 * HS_STRIDE + k]);
                v2f b;
                b.x = Vp[(size_t)k       * DOUT + ncol];
                b.y = Vp[(size_t)(k + 1) * DOUT + ncol];
                acc = __builtin_amdgcn_wmma_f32_16x16x4_f32(
                    false, a, false, b, (short)0, acc, false, false);
            }
            float bv = bias[ncol];
            #pragma unroll
            for (int j = 0; j < 8; ++j) {
                int m  = j + 8 * lhalf;
                int gm = t0 + m;
                if (gm < cnt)
                    out[(size_t)toks[m] * DOUT + ncol] = acc[j] + bv;
            }
        }
    }
}

// ---------------- host-side launch ----------------

extern "C" void kernel_launch(void* const* d_in, const int* in_sizes, int n_in,
                              void* d_out, int out_size, void* d_ws, size_t ws_size,
                              hipStream_t stream) {
    const float* x    = (const float*)d_in[0];
    const int*   pids = (const int*)  d_in[1];
    const float* U    = (const float*)d_in[2];
    const float* V    = (const float*)d_in[3];
    const float* bias = (const float*)d_in[4];
    float*       out  = (float*)d_out;

    int* ws      = (int*)d_ws;
    int* counts  = ws;            // [64]
    int* cursor  = ws + 64;       // [64]
    int* offsets = ws + 128;      // [65]
    int* ntiles  = ws + 200;      // [1]
    int* tiles   = ws + 256;      // [MAX_TILES = 320]
    int* order   = ws + 1024;     // [4096]

    k_zero   <<<1, 64, 0, stream>>>(counts, cursor);
    k_count  <<<NTOK / 256, 256, 0, stream>>>(pids, counts);
    k_scan   <<<1, 32, 0, stream>>>(counts, offsets, tiles, ntiles);
    k_scatter<<<NTOK / 256, 256, 0, stream>>>(pids, offsets, cursor, order);

    // Exactly MAX_TILES blocks; those past the runtime tile count exit on one compare.
    grouped_lowrank_gemm<<<MAX_TILES, 128, 0, stream>>>(
        x, U, V, bias, offsets, order, tiles, ntiles, out);
}